// RNN_39822936768618
// MI455X (gfx1250) — compile-verified
//
#include <hip/hip_runtime.h>
#include <math.h>

typedef __attribute__((ext_vector_type(2))) float v2f;
typedef __attribute__((ext_vector_type(8))) float v8f;

#define B_ 8192
#define T_ 2048
#define I_ 2
#define H_ 20
#define KP 24      // augmented K: 20 hidden + 2 input + 1 bias + 1 zero pad
#define NCHUNK 6   // KP / 4  (K-chunks of 4 for v_wmma_f32_16x16x4_f32)
#define STRIDE 28  // LDS row stride (floats): 8B-aligned b64 loads, conflict-free stores

// Branch-free tanh on the hardware TRANS pipe:
//   tanh(x) = 1 - 2/(2^(2*log2e*x)+1)
// v_exp_f32 saturates to +inf / 0 at range ends, pinning the result to +/-1.
__device__ __forceinline__ float fast_tanh(float x) {
#if __has_builtin(__builtin_amdgcn_exp2f)
    float e = __builtin_amdgcn_exp2f(x * 2.8853900817779268f); // 2*log2(e)
#else
    float e = __expf(x + x);
#endif
    return fmaf(-2.0f, __builtin_amdgcn_rcpf(e + 1.0f), 1.0f);
}

// W_aug[k][n] for the augmented recurrence GEMM:
//   pre[b,n] = sum_k h_aug[b,k] * W_aug[k][n]
//            = (h @ W_hh^T)[b,n] + x0*W_ih[n,0] + x1*W_ih[n,1] + b_ih[n] + b_hh[n]
__device__ __forceinline__ float waug(int k, int n,
                                      const float* __restrict__ W_ih,
                                      const float* __restrict__ W_hh,
                                      const float* __restrict__ b_ih,
                                      const float* __restrict__ b_hh) {
    if (n >= H_) return 0.0f;
    if (k < H_) return W_hh[n * H_ + k];
    if (k == H_)     return W_ih[n * I_ + 0];
    if (k == H_ + 1) return W_ih[n * I_ + 1];
    if (k == H_ + 2) return b_ih[n] + b_hh[n];
    return 0.0f; // k == 23 pad
}

__global__ __launch_bounds__(32) void rnn_scan_wmma(
    const float* __restrict__ x,
    const float* __restrict__ W_ih,
    const float* __restrict__ W_hh,
    const float* __restrict__ b_ih,
    const float* __restrict__ b_hh,
    const float* __restrict__ fc_w,
    const float* __restrict__ fc_b,
    float* __restrict__ out)
{
    __shared__ float hbuf[16 * STRIDE]; // per-wave h_aug tile: [16 batch][KP(+pad)]
    // NOTE: slots 16..19 of each row hold PRE-activation values (tanh is applied
    // on the A-operand side for K-chunk 4, and in the epilogue).

    const int lane = threadIdx.x;      // 0..31, wave32
    const int base = blockIdx.x * 16;  // batch-tile start
    const int n0 = lane & 15;
    const int khalf = (lane >> 4) * 2; // lanes 0-15 -> K pair k0,k0+1 ; 16-31 -> k0+2,k0+3

    // ---- Build W_aug in registers, WMMA-B layout (mirrors documented A 16x4 layout):
    //   lanes 0-15:  N=lane,    vgpr0=K(4c),   vgpr1=K(4c+1)
    //   lanes 16-31: N=lane-16, vgpr0=K(4c+2), vgpr1=K(4c+3)
    v2f Bop[2][NCHUNK];
    #pragma unroll
    for (int tile = 0; tile < 2; ++tile) {
        const int n = n0 + 16 * tile;
        #pragma unroll
        for (int c = 0; c < NCHUNK; ++c) {
            const int k0 = 4 * c + khalf;
            Bop[tile][c].x = waug(k0,     n, W_ih, W_hh, b_ih, b_hh);
            Bop[tile][c].y = waug(k0 + 1, n, W_ih, W_hh, b_ih, b_hh);
        }
    }

    // ---- Init h_aug: h0 = 0, slot 22 = 1.0 (bias lane), slot 23 = 0
    for (int idx = lane; idx < 16 * STRIDE; idx += 32) hbuf[idx] = 0.0f;
    __syncthreads(); // single-wave WG: cheap; orders init vs. loop
    if (lane < 16) hbuf[lane * STRIDE + (H_ + 2)] = 1.0f;

    // Per-lane x pointer; lanes 16-31 duplicate lanes 0-15 (same cachelines, no
    // extra traffic) so the prefetch load needs no EXEC manipulation.
    const float* xp = x + (long)(base + n0) * T_ * I_;
    const int arow = n0 * STRIDE + khalf; // A-operand load row (float index), 8B aligned
    const int xslot = n0 * STRIDE + H_;   // K slots 20,21

    v2f xv = *(const v2f*)xp; // prefetch x_0
    xp += I_;

    for (int t = 0; t < T_; ++t) {
        // Feed x_t into augmented K slots 20,21. Upper-half lanes write the same
        // address/data as their lower-half twin (benign duplicate store).
        *(v2f*)&hbuf[xslot] = xv;

        // Augmented GEMM: [16 x 24] x [24 x 32], fp32 WMMA, 6 K-chunks x 2 N-tiles.
        // Chunk 4 (k=16..19) carries the tile-1 feedback stored pre-activation:
        // apply tanh to just those 2 values/lane (the only useful tile-1 outputs).
        v8f acc0 = {};
        v8f acc1 = {};
        #pragma unroll
        for (int c = 0; c < NCHUNK; ++c) {
            v2f a = *(const v2f*)&hbuf[arow + 4 * c]; // in-order LDS vs. stores above
            if (c == 4) {
                a.x = fast_tanh(a.x);
                a.y = fast_tanh(a.y);
            }
            acc0 = __builtin_amdgcn_wmma_f32_16x16x4_f32(
                false, a, false, Bop[0][c], (short)0, acc0, false, false);
            acc1 = __builtin_amdgcn_wmma_f32_16x16x4_f32(
                false, a, false, Bop[1][c], (short)0, acc1, false, false);
        }

        // Prefetch x_{t+1} (uniform guard -> loop peel, no exec masking); the
        // load latency hides behind the activation + scatter tail.
        if (t + 1 < T_) xv = *(const v2f*)xp;
        xp += I_;

        // Branch-free activation for tile 0 (n=0..15) only.
        #pragma unroll
        for (int r = 0; r < 8; ++r) acc0[r] = fast_tanh(acc0[r]);

        // Scatter h_{t+1} back to h_aug (C/D layout -> A layout via LDS).
        // Tile0: post-tanh, all 16 columns. Tile1: raw, only n=16..19 (predicate
        // keeps slots 20..23 = x / bias / pad intact).
        const int mbase = (lane >> 4) * 8; // lanes 0-15: rows 0..7 ; 16-31: rows 8..15
        #pragma unroll
        for (int r = 0; r < 8; ++r)
            hbuf[(mbase + r) * STRIDE + n0] = acc0[r];
        if (n0 < (H_ - 16)) {
            #pragma unroll
            for (int r = 0; r < 8; ++r)
                hbuf[(mbase + r) * STRIDE + 16 + n0] = acc1[r];
        }
    }

    // ---- Epilogue: h_state = h_T, out = h_T @ fc_w^T + fc_b
    // Slots 16..19 are pre-activation -> apply tanh here.
    for (int idx = lane; idx < 16 * H_; idx += 32) {
        const int m = idx / H_, hh = idx % H_;
        float v = hbuf[m * STRIDE + hh];
        if (hh >= 16) v = fast_tanh(v);
        out[B_ + (long)(base + m) * H_ + hh] = v;
    }
    if (lane < 16) {
        float acc = fc_b[0];
        #pragma unroll
        for (int hh = 0; hh < H_; ++hh) {
            float v = hbuf[lane * STRIDE + hh];
            if (hh >= 16) v = fast_tanh(v);
            acc += v * fc_w[hh];
        }
        out[base + lane] = acc;
    }
}

extern "C" void kernel_launch(void* const* d_in, const int* in_sizes, int n_in,
                              void* d_out, int out_size, void* d_ws, size_t ws_size,
                              hipStream_t stream) {
    const float* x    = (const float*)d_in[0];
    const float* W_ih = (const float*)d_in[1];
    const float* W_hh = (const float*)d_in[2];
    const float* b_ih = (const float*)d_in[3];
    const float* b_hh = (const float*)d_in[4];
    const float* fc_w = (const float*)d_in[5];
    const float* fc_b = (const float*)d_in[6];
    float* out = (float*)d_out;

    rnn_scan_wmma<<<B_ / 16, 32, 0, stream>>>(x, W_ih, W_hh, b_ih, b_hh,
                                              fc_w, fc_b, out);
    (void)in_sizes; (void)n_in; (void)out_size; (void)d_ws; (void)ws_size;
}